// Measurements_74878459838521
// MI455X (gfx1250) — compile-verified
//
#include <hip/hip_runtime.h>

// MI455X (gfx1250) — memory-bound fused pad+einsum+patch-extract.
// out[b,h,w] = sum_t x[b,t,h,w] * weight[t, h%4, w%4]  (512x512 padded plane)
// Chained V_WMMA_F32_16X16X4_F32 (K=16 over t), weights in A (shared operand):
//   A[m,t] = weight[t, m/4, m%4] = wgt[t*16 + m]   (loop-invariant, 8 regs/lane)
//   B[t,p] = x[t, pixel p]                          (16 consecutive-w pixels)
//   D[m,p] = pixel p's output at coset m; keep m = 4*(h%4) + p%4.
// D layout is column(=pixel)-per-lane, so each lane owns its own pixel's
// result in a uniform half at component 4*(h%4&1) + (lane&3): branch-free
// extraction, no LDS, coalesced 16-lane stores.

typedef __attribute__((ext_vector_type(2))) float v2f;
typedef __attribute__((ext_vector_type(8))) float v8f;

#define H_IN 500
#define W_IN 500
#define T_IN 16
#define HP   512
#define WP   512
#define NP   127
#define TSTR (H_IN * W_IN)       // t-stride in input = 250000
#define NPATCH_ELEMS (8 * NP * NP * 64)

__global__ __launch_bounds__(256) void einsum_wmma_kernel(
    const float* __restrict__ x,      // (8,1,16,500,500)
    const float* __restrict__ wgt,    // (16,4,4) row-major == wgt[t*16 + coset]
    float* __restrict__ full,         // ws: (8,512,512)
    float* __restrict__ crop)         // d_out tail: (8,500,500)
{
    const int lane = threadIdx.x & 31;
    const int wave = threadIdx.x >> 5;
    const int half = lane >> 4;       // K pair {2h, 2h+1} within each x4 step
    const int n    = lane & 15;       // A row (coset) / B column (pixel)
    const int b    = blockIdx.z;
    const int h    = blockIdx.y;

    const size_t fullbase = ((size_t)b * HP + h) * WP;

    if (h >= H_IN) {                  // uniform pad row -> zeros, no loads
        full[fullbase + threadIdx.x]       = 0.f;
        full[fullbase + 256 + threadIdx.x] = 0.f;
        return;
    }

    // ---- A operand (weights): invariant; 8 floats per lane ----
    const int k0 = 2 * half;
    v2f a0, a1, a2, a3;
    a0.x = wgt[( 0 + k0) * 16 + n];  a0.y = wgt[( 0 + k0 + 1) * 16 + n];
    a1.x = wgt[( 4 + k0) * 16 + n];  a1.y = wgt[( 4 + k0 + 1) * 16 + n];
    a2.x = wgt[( 8 + k0) * 16 + n];  a2.y = wgt[( 8 + k0 + 1) * 16 + n];
    a3.x = wgt[(12 + k0) * 16 + n];  a3.y = wgt[(12 + k0 + 1) * 16 + n];

    const int r        = h & 3;       // uniform h coset
    const int res_half = r >> 1;      // lane-half holding result rows
    const int base_c   = (r & 1) * 4; // component base within that half
    const int c2       = lane & 3;

    const float*  xrow     = x + (size_t)b * T_IN * TSTR + (size_t)h * W_IN;
    const size_t  cropbase = ((size_t)b * H_IN + h) * W_IN;

    // Each wave covers 64 pixels (4 chains of 16); block = one 512-wide row.
    #pragma unroll
    for (int chain = 0; chain < 4; ++chain) {
        const int  w0   = wave * 64 + chain * 16;
        const int  w    = w0 + n;
        const bool inw  = w < W_IN;
        const int  wsc  = inw ? w : (W_IN - 1);   // clamped: loads unconditional

        // B operand: x[t, pixel n]; zero whole column via value select (no branch)
        const float* xp = xrow + wsc;
        const float l0 = xp[( 0 + k0) * TSTR], l1 = xp[( 1 + k0) * TSTR];
        const float l2 = xp[( 4 + k0) * TSTR], l3 = xp[( 5 + k0) * TSTR];
        const float l4 = xp[( 8 + k0) * TSTR], l5 = xp[( 9 + k0) * TSTR];
        const float l6 = xp[(12 + k0) * TSTR], l7 = xp[(13 + k0) * TSTR];
        v2f b0, b1, b2, b3;
        b0.x = inw ? l0 : 0.f;  b0.y = inw ? l1 : 0.f;
        b1.x = inw ? l2 : 0.f;  b1.y = inw ? l3 : 0.f;
        b2.x = inw ? l4 : 0.f;  b2.y = inw ? l5 : 0.f;
        b3.x = inw ? l6 : 0.f;  b3.y = inw ? l7 : 0.f;

        v8f acc = {0.f, 0.f, 0.f, 0.f, 0.f, 0.f, 0.f, 0.f};
        acc = __builtin_amdgcn_wmma_f32_16x16x4_f32(false, a0, false, b0, (short)0, acc, false, false);
        acc = __builtin_amdgcn_wmma_f32_16x16x4_f32(false, a1, false, b1, (short)0, acc, false, false);
        acc = __builtin_amdgcn_wmma_f32_16x16x4_f32(false, a2, false, b2, (short)0, acc, false, false);
        acc = __builtin_amdgcn_wmma_f32_16x16x4_f32(false, a3, false, b3, (short)0, acc, false, false);

        // Branch-free extraction: lane owns pixel n; pick component base_c + c2.
        float val;
        if (base_c) {                 // uniform
            val = acc[4];
            val = (c2 == 1) ? acc[5] : val;
            val = (c2 == 2) ? acc[6] : val;
            val = (c2 == 3) ? acc[7] : val;
        } else {
            val = acc[0];
            val = (c2 == 1) ? acc[1] : val;
            val = (c2 == 2) ? acc[2] : val;
            val = (c2 == 3) ? acc[3] : val;
        }

        if (half == res_half) {       // 16 active lanes, consecutive addresses
            full[fullbase + w] = val;
            if (inw) crop[cropbase + w] = val;
        }
    }
}

// patches[b, ph*127+pw, i*8+j] = full[b, ph*4 + j, pw*4 + i]
__global__ __launch_bounds__(256) void patches_kernel(
    const float* __restrict__ full, float* __restrict__ out)
{
    const int idx = blockIdx.x * 256 + threadIdx.x;
    if (idx >= NPATCH_ELEMS) return;
    const int e  = idx & 63;
    const int p  = (idx >> 6) % (NP * NP);
    const int b  = idx / (64 * NP * NP);
    const int i  = e >> 3;            // w offset inside patch
    const int j  = e & 7;             // h offset inside patch
    const int ph = p / NP;
    const int pw = p % NP;
    const int h  = ph * 4 + j;
    const int w  = pw * 4 + i;
    out[idx] = full[((size_t)b * HP + h) * WP + w];
}

extern "C" void kernel_launch(void* const* d_in, const int* in_sizes, int n_in,
                              void* d_out, int out_size, void* d_ws, size_t ws_size,
                              hipStream_t stream) {
    const float* x   = (const float*)d_in[0];   // input  (8,1,16,500,500) f32
    const float* wgt = (const float*)d_in[1];   // weight (16,4,4) f32
    float* out   = (float*)d_out;               // patches (8,16129,64) ...
    float* crop  = out + NPATCH_ELEMS;          // ... then cropped (8,500,500)
    float* full  = (float*)d_ws;                // 8*512*512*4 = 8 MB scratch

    dim3 g1(1, HP, 8);                          // one 512-wide row per block
    einsum_wmma_kernel<<<g1, 256, 0, stream>>>(x, wgt, full, crop);

    const int nblk = (NPATCH_ELEMS + 255) / 256;
    patches_kernel<<<nblk, 256, 0, stream>>>(full, out);
}